// Memory_70093866270800
// MI455X (gfx1250) — compile-verified
//
#include <hip/hip_runtime.h>
#include <math.h>

constexpr int B_ = 16;
constexpr int N_ = 2048;
constexpr int WC_ = 64;
constexpr int R_ = 4;
constexpr int IN_ = 512;
constexpr int CTRL_ = 471;   // total controller output columns
constexpr int CTRLP_ = 480;  // padded to multiple of 16
constexpr float DELTA_ = 1e-6f;

// ---- workspace layout (float elements) ----
constexpr int WS_WCAT  = 0;                       // 480*512 packed weights (row n = concat weight row)
constexpr int WS_BCAT  = WS_WCAT + CTRLP_ * IN_;  // 480 packed bias
constexpr int WS_Z     = WS_BCAT + CTRLP_;        // 16*480 raw projections
constexpr int WS_ACT   = WS_Z + B_ * CTRLP_;      // 16*480 activated projections
constexpr int WS_MNO   = WS_ACT + B_ * CTRLP_;    // 16*2048 old memory row norms
constexpr int WS_WCWS  = WS_MNO + B_ * N_;        // 16*2048 write-content sim (scaled)
constexpr int WS_WCW   = WS_WCWS + B_ * N_;       // 16*2048 write-content weighting
constexpr int WS_ALLOC = WS_WCW + B_ * N_;        // 16*2048 allocation weighting
constexpr int WS_MNN   = WS_ALLOC + B_ * N_;      // 16*2048 new memory row norms
constexpr int WS_RCWS  = WS_MNN + B_ * N_;        // 16*4*2048 read-content sim (scaled)
constexpr int WS_RCW   = WS_RCWS + B_ * R_ * N_;  // 16*4*2048 read-content weighting
constexpr int WS_FWD   = WS_RCW + B_ * R_ * N_;   // 16*4*2048 forward weighting
constexpr int WS_BWD   = WS_FWD + B_ * R_ * N_;   // 16*4*2048 backward weighting (atomically accumulated)

// ---- output layout (tuple concatenated flat, float elements) ----
constexpr size_t OUT_RV    = 0;                                  // (16,4,64)
constexpr size_t OUT_MEM   = OUT_RV + (size_t)B_ * R_ * WC_;     // (16,2048,64)
constexpr size_t OUT_LINK  = OUT_MEM + (size_t)B_ * N_ * WC_;    // (16,1,2048,2048)
constexpr size_t OUT_PREC  = OUT_LINK + (size_t)B_ * N_ * N_;    // (16,1,2048)
constexpr size_t OUT_RW    = OUT_PREC + (size_t)B_ * N_;         // (16,4,2048)
constexpr size_t OUT_WW    = OUT_RW + (size_t)B_ * R_ * N_;      // (16,1,2048)
constexpr size_t OUT_USAGE = OUT_WW + (size_t)B_ * N_;           // (16,2048)

typedef float v2f __attribute__((ext_vector_type(2)));
typedef float v8f __attribute__((ext_vector_type(8)));

// CDNA5 f32 WMMA: D(16x16) = A(16x4) * B(4x16) + C, exact fp32.
// A: lanes 0-15 hold rows M=lane {K=k0,k0+1}; lanes 16-31 hold rows M=lane-16 {K=k0+2,k0+3}.
// B: lanes 0-15 hold cols N=lane {K=k0,k0+1}; lanes 16-31 hold cols N=lane-16 {K=k0+2,k0+3}.
// C/D: VGPR v -> (M=v, N=lane) for lanes 0-15; (M=v+8, N=lane-16) for lanes 16-31.
__device__ inline v8f wmma_f32_16x16x4(v2f a, v2f b, v8f c) {
  return __builtin_amdgcn_wmma_f32_16x16x4_f32(false, a, false, b, (short)0, c, false, false);
}

__device__ inline float sigmoidf_(float x) { return 1.0f / (1.0f + expf(-x)); }
__device__ inline float softplusf_(float x) { return (x > 20.0f) ? x : log1pf(expf(x)); }

// -------------------- 0: misc --------------------
__global__ void k_zero(float* p, int n) {
  int i = blockIdx.x * blockDim.x + threadIdx.x;
  if (i < n) p[i] = 0.0f;
}

// -------------------- 1: pack all controller weights into one 480x512 matrix --------------------
__global__ void k_pack(const float* Wrk, const float* brk, const float* Wrs, const float* brs,
                       const float* Wwk, const float* bwk, const float* Wws, const float* bws,
                       const float* Wev, const float* bev, const float* Wwv, const float* bwv,
                       const float* Wfg, const float* bfg, const float* Wag, const float* bag,
                       const float* Wwg, const float* bwg, const float* Wrm, const float* brm,
                       float* ws) {
  int idx = blockIdx.x * blockDim.x + threadIdx.x;
  if (idx >= CTRLP_ * IN_) return;
  int n = idx / IN_, k = idx % IN_;
  const float* W = nullptr; const float* bb = nullptr; int row = 0;
  if (n < 256)      { W = Wrk; bb = brk; row = n; }
  else if (n < 260) { W = Wrs; bb = brs; row = n - 256; }
  else if (n < 324) { W = Wwk; bb = bwk; row = n - 260; }
  else if (n < 325) { W = Wws; bb = bws; row = 0; }
  else if (n < 389) { W = Wev; bb = bev; row = n - 325; }
  else if (n < 453) { W = Wwv; bb = bwv; row = n - 389; }
  else if (n < 457) { W = Wfg; bb = bfg; row = n - 453; }
  else if (n < 458) { W = Wag; bb = bag; row = 0; }
  else if (n < 459) { W = Wwg; bb = bwg; row = 0; }
  else if (n < 471) { W = Wrm; bb = brm; row = n - 459; }
  ws[WS_WCAT + n * IN_ + k] = W ? W[row * IN_ + k] : 0.0f;
  if (k == 0) ws[WS_BCAT + n] = W ? bb[row] : 0.0f;
}

// -------------------- 2: controller GEMM  z(16x480) = x(16x512) @ Wcat^T  (WMMA f32) ----------
__global__ void k_ctrl_gemm(const float* __restrict__ x, float* __restrict__ ws) {
  const float* wcat = ws + WS_WCAT;
  float* z = ws + WS_Z;
  const int lane = threadIdx.x;            // exactly one wave32 per block
  const int n0 = blockIdx.x * 16;
  const int row = lane & 15;
  const bool hi = lane >= 16;
  v8f acc = {};
  for (int k0 = 0; k0 < IN_; k0 += 4) {
    const int ka = k0 + (hi ? 2 : 0);
    v2f a, b;
    a.x = x[row * IN_ + ka];     a.y = x[row * IN_ + ka + 1];
    const int n = n0 + row;
    b.x = wcat[n * IN_ + ka];    b.y = wcat[n * IN_ + ka + 1];
    acc = wmma_f32_16x16x4(a, b, acc);
  }
  const int m0 = hi ? 8 : 0;
  const int nn = n0 + row;
  for (int v = 0; v < 8; ++v) z[(m0 + v) * CTRLP_ + nn] = acc[v];
}

// -------------------- 3: bias + per-column activation --------------------
__global__ void k_act(float* ws) {
  int idx = blockIdx.x * blockDim.x + threadIdx.x;
  if (idx >= B_ * CTRL_) return;
  int b = idx / CTRL_, c = idx % CTRL_;
  const float* z = ws + WS_Z;
  const float* bc = ws + WS_BCAT;
  float raw = z[b * CTRLP_ + c] + bc[c];
  float o;
  if (c < 256)      o = tanhf(raw);       // read_keys
  else if (c < 260) o = softplusf_(raw);  // read_strengths
  else if (c < 324) o = tanhf(raw);       // write_key
  else if (c < 325) o = softplusf_(raw);  // write_strength
  else if (c < 389) o = sigmoidf_(raw);   // erase_vector
  else if (c < 453) o = tanhf(raw);       // write_vector
  else if (c < 459) o = sigmoidf_(raw);   // free_gates, alloc_gate, write_gate
  else {                                  // read_modes: softmax over groups of 3
    int g0 = 459 + ((c - 459) / 3) * 3;
    float a0 = z[b * CTRLP_ + g0] + bc[g0];
    float a1 = z[b * CTRLP_ + g0 + 1] + bc[g0 + 1];
    float a2 = z[b * CTRLP_ + g0 + 2] + bc[g0 + 2];
    float m = fmaxf(a0, fmaxf(a1, a2));
    float s = expf(a0 - m) + expf(a1 - m) + expf(a2 - m);
    o = expf(raw - m) / s;
  }
  ws[WS_ACT + b * CTRLP_ + c] = o;
}

// -------------------- 4: usage update + retention --------------------
__global__ void k_usage(const float* __restrict__ uv, const float* __restrict__ wwrite,
                        const float* __restrict__ rw, const float* __restrict__ ws,
                        float* __restrict__ out_usage) {
  int idx = blockIdx.x * blockDim.x + threadIdx.x;
  if (idx >= B_ * N_) return;
  int b = idx >> 11, n = idx & (N_ - 1);
  const float* act = ws + WS_ACT;
  float u = uv[idx];
  float u1 = u + (1.0f - u) * wwrite[idx];   // H=1 -> 1-prod(1-ww) == ww
  float ret = 1.0f;
  for (int r = 0; r < R_; ++r)
    ret *= 1.0f - act[b * CTRLP_ + 453 + r] * rw[(b * R_ + r) * N_ + n];
  out_usage[idx] = u1 * ret;
}

// -------------------- 5: row norms of a (rows x 64) matrix, one wave per row ----------
__global__ void k_rownorm(const float* __restrict__ src, float* __restrict__ dst, int rows) {
  int wave = threadIdx.x >> 5, lane = threadIdx.x & 31;
  int row = blockIdx.x * 8 + wave;
  if (row >= rows) return;
  float v0 = src[row * WC_ + lane];
  float v1 = src[row * WC_ + 32 + lane];
  float ss = v0 * v0 + v1 * v1;
  for (int m = 16; m >= 1; m >>= 1) ss += __shfl_xor(ss, m, 32);
  if (lane == 0) dst[row] = sqrtf(ss);
}

// -------------------- 6: write-key content sim (scaled by strength) --------------------
__global__ void k_wcw_sim(const float* __restrict__ memory, float* __restrict__ ws) {
  int idx = blockIdx.x * blockDim.x + threadIdx.x;
  if (idx >= B_ * N_) return;
  int b = idx >> 11, n = idx & (N_ - 1);
  const float* act = ws + WS_ACT;
  const float* wk = act + b * CTRLP_ + 260;
  float kn = 0.0f, dot = 0.0f;
  for (int w = 0; w < WC_; ++w) {
    float kv = wk[w];
    kn += kv * kv;
    dot += kv * memory[(size_t)(b * N_ + n) * WC_ + w];
  }
  float mn = ws[WS_MNO + idx];
  float str = act[b * CTRLP_ + 324];
  ws[WS_WCWS + idx] = dot / (sqrtf(kn) * mn + DELTA_) * str;
}

// -------------------- 7: row softmax (len per row, one workgroup per row) ----------
__global__ void k_softmax(const float* __restrict__ src, float* __restrict__ dst, int len) {
  __shared__ float red[256];
  int row = blockIdx.x, t = threadIdx.x;
  const float* s = src + (size_t)row * len;
  float* d = dst + (size_t)row * len;
  float vals[8];
  int cnt = 0;
  float mx = -1e30f;
  for (int i = t; i < len; i += 256) { vals[cnt] = s[i]; mx = fmaxf(mx, vals[cnt]); ++cnt; }
  red[t] = mx; __syncthreads();
  for (int o = 128; o > 0; o >>= 1) { if (t < o) red[t] = fmaxf(red[t], red[t + o]); __syncthreads(); }
  mx = red[0]; __syncthreads();
  float sum = 0.0f;
  for (int q = 0; q < cnt; ++q) { vals[q] = expf(vals[q] - mx); sum += vals[q]; }
  red[t] = sum; __syncthreads();
  for (int o = 128; o > 0; o >>= 1) { if (t < o) red[t] += red[t + o]; __syncthreads(); }
  float inv = 1.0f / red[0];
  int q = 0;
  for (int i = t; i < len; i += 256) d[i] = vals[q++] * inv;
}

// -------------------- 8: allocation (bitonic argsort + prefix product in LDS) ----------
__global__ void __launch_bounds__(256) k_alloc(const float* __restrict__ usage, float* __restrict__ ws) {
  __shared__ float su[N_];
  __shared__ int sidx[N_];
  __shared__ float pp[N_];
  int b = blockIdx.x, t = threadIdx.x;
  for (int i = t; i < N_; i += 256) {
    su[i] = DELTA_ + (1.0f - DELTA_) * usage[b * N_ + i];
    sidx[i] = i;
  }
  __syncthreads();
  // bitonic sort ascending
  for (int k = 2; k <= N_; k <<= 1) {
    for (int j = k >> 1; j > 0; j >>= 1) {
      for (int i = t; i < N_; i += 256) {
        int ij = i ^ j;
        if (ij > i) {
          bool up = ((i & k) == 0);
          float a = su[i], c = su[ij];
          if ((a > c) == up) {
            su[i] = c; su[ij] = a;
            int tmp = sidx[i]; sidx[i] = sidx[ij]; sidx[ij] = tmp;
          }
        }
      }
      __syncthreads();
    }
  }
  // inclusive prefix product (Hillis-Steele)
  for (int i = t; i < N_; i += 256) pp[i] = su[i];
  __syncthreads();
  for (int off = 1; off < N_; off <<= 1) {
    float tmp[8]; int s = 0;
    for (int i = t; i < N_; i += 256, ++s) tmp[s] = (i >= off) ? pp[i - off] : 1.0f;
    __syncthreads();
    s = 0;
    for (int i = t; i < N_; i += 256, ++s) pp[i] *= tmp[s];
    __syncthreads();
  }
  // alloc[orig_index] = (1 - sorted_u) * exclusive_prefix_prod
  for (int i = t; i < N_; i += 256) {
    float excl = (i == 0) ? 1.0f : pp[i - 1];
    ws[WS_ALLOC + b * N_ + sidx[i]] = (1.0f - su[i]) * excl;
  }
}

// -------------------- 9: write weighting + precedence --------------------
__global__ void k_ww_prec(const float* __restrict__ prec_old, float* __restrict__ ws,
                          float* __restrict__ out) {
  __shared__ float red[256];
  int b = blockIdx.x, t = threadIdx.x;
  const float* act = ws + WS_ACT;
  float ag = act[b * CTRLP_ + 457], wg = act[b * CTRLP_ + 458];
  float* ww_out = out + OUT_WW + (size_t)b * N_;
  float lsum = 0.0f;
  for (int i = t; i < N_; i += 256) {
    float w = wg * (ag * ws[WS_ALLOC + b * N_ + i] + (1.0f - ag) * ws[WS_WCW + b * N_ + i]);
    ww_out[i] = w;
    lsum += w;
  }
  red[t] = lsum; __syncthreads();
  for (int o = 128; o > 0; o >>= 1) { if (t < o) red[t] += red[t + o]; __syncthreads(); }
  float S = red[0];
  for (int i = t; i < N_; i += 256)
    out[OUT_PREC + (size_t)b * N_ + i] = (1.0f - S) * prec_old[b * N_ + i] + ww_out[i];
}

// -------------------- 10: memory update --------------------
__global__ void k_memnew(const float* __restrict__ memory, const float* __restrict__ ws,
                         float* __restrict__ out) {
  int idx = blockIdx.x * blockDim.x + threadIdx.x;
  if (idx >= B_ * N_ * WC_) return;
  int b = idx / (N_ * WC_);
  int n = (idx / WC_) & (N_ - 1);
  int w = idx & (WC_ - 1);
  float wwv = out[OUT_WW + (size_t)b * N_ + n];
  const float* act = ws + WS_ACT;
  float er = act[b * CTRLP_ + 325 + w];
  float wv = act[b * CTRLP_ + 389 + w];
  out[OUT_MEM + idx] = memory[idx] * (1.0f - wwv * er) + wwv * wv;
}

// -------------------- 11: read content sim via WMMA f32 (read_keys @ mem_new^T) ----------
__global__ void k_rcw_sim(const float* __restrict__ memnew, float* __restrict__ ws) {
  const int lane = threadIdx.x;            // one wave32 per block
  const int b = blockIdx.x >> 7;           // 128 n-tiles per batch
  const int nt = blockIdx.x & 127;
  const int n0 = nt * 16;
  const float* act = ws + WS_ACT;
  const int row = lane & 15;
  const bool hi = lane >= 16;
  v8f acc = {};
  for (int k0 = 0; k0 < WC_; k0 += 4) {
    const int ka = k0 + (hi ? 2 : 0);
    v2f a, bv;
    if (row < R_) { a.x = act[b * CTRLP_ + row * WC_ + ka]; a.y = act[b * CTRLP_ + row * WC_ + ka + 1]; }
    else          { a.x = 0.0f; a.y = 0.0f; }
    const int n = n0 + row;
    bv.x = memnew[(size_t)(b * N_ + n) * WC_ + ka];
    bv.y = memnew[(size_t)(b * N_ + n) * WC_ + ka + 1];
    acc = wmma_f32_16x16x4(a, bv, acc);
  }
  if (lane < 16) {
    float mn = ws[WS_MNN + b * N_ + n0 + lane];
    for (int v = 0; v < R_; ++v) {
      const float* rk = act + b * CTRLP_ + v * WC_;
      float kn = 0.0f;
      for (int w = 0; w < WC_; ++w) kn += rk[w] * rk[w];
      float str = act[b * CTRLP_ + 256 + v];
      ws[WS_RCWS + (b * R_ + v) * N_ + n0 + lane] = acc[v] / (sqrtf(kn) * mn + DELTA_) * str;
    }
  }
}

// -------------------- 12: fused link update + fwd (exact) + bwd (atomics) ----------
// One block per (b, i-tile of 64). Streams all j, LDS-staged 64x64 link tiles.
__global__ void __launch_bounds__(256) k_link(const float* __restrict__ LM,
                                              const float* __restrict__ prec,
                                              const float* __restrict__ rw_old,
                                              float* __restrict__ ws,
                                              float* __restrict__ out) {
  __shared__ float tile[64][65];
  __shared__ float rwj[R_][64];
  __shared__ float rwi[R_][64];
  const int t = threadIdx.x;
  const int b = blockIdx.x >> 5;
  const int it = blockIdx.x & 31;
  const int i0 = it * 64;
  const float* wwp = out + OUT_WW + (size_t)b * N_;
  const int ig = t >> 6;     // 0..3
  const int jl = t & 63;     // 0..63
  // preload rw over this block's i range, and ww for the 16 rows this thread owns
  rwi[ig][jl] = rw_old[(b * R_ + ig) * N_ + i0 + jl];
  float wwi_r[16];
  for (int s = 0; s < 16; ++s) wwi_r[s] = wwp[i0 + s * 4 + ig];
  float fwdsum = 0.0f;
  for (int jt = 0; jt < 32; ++jt) {
    __syncthreads();  // previous tile fully consumed
    const int j0 = jt * 64;
    rwj[ig][jl] = rw_old[(b * R_ + ig) * N_ + j0 + jl];
    const int j = j0 + jl;
    const float wwj = wwp[j];
    const float pj = prec[(size_t)b * N_ + j];
    for (int s = 0; s < 16; ++s) {
      const int il = s * 4 + ig;
      const int i = i0 + il;
      const float lm = LM[((size_t)b * N_ + i) * N_ + j];
      float lk = (1.0f - wwi_r[s] - wwj) * lm + wwi_r[s] * pj;
      if (i == j) lk = 0.0f;
      out[OUT_LINK + ((size_t)b * N_ + i) * N_ + j] = lk;
      tile[il][jl] = lk;
    }
    __syncthreads();
    // fwd: this thread owns (r=ig, i_local=jl); accumulate over the tile's j
    float fa = 0.0f;
    for (int q = 0; q < 64; ++q) fa += tile[jl][q] * rwj[ig][q];
    fwdsum += fa;
    // bwd: this thread owns (r=ig, j_local=jl); partial over this i-tile -> atomic
    float ba = 0.0f;
    for (int q = 0; q < 64; ++q) ba += tile[q][jl] * rwi[ig][q];
    atomicAdd(&ws[WS_BWD + (b * R_ + ig) * N_ + j0 + jl], ba);
  }
  ws[WS_FWD + (b * R_ + ig) * N_ + i0 + jl] = fwdsum;  // exact, no atomic
}

// -------------------- 13: combine read modes --------------------
__global__ void k_rwnew(const float* __restrict__ ws, float* __restrict__ out) {
  int idx = blockIdx.x * blockDim.x + threadIdx.x;
  if (idx >= B_ * R_ * N_) return;
  int b = idx / (R_ * N_);
  int r = (idx / N_) & (R_ - 1);
  const float* act = ws + WS_ACT;
  float m0 = act[b * CTRLP_ + 459 + r * 3 + 0];
  float m1 = act[b * CTRLP_ + 459 + r * 3 + 1];
  float m2 = act[b * CTRLP_ + 459 + r * 3 + 2];
  out[OUT_RW + idx] = m0 * ws[WS_BWD + idx] + m1 * ws[WS_FWD + idx] + m2 * ws[WS_RCW + idx];
}

// -------------------- 14: read vectors via WMMA f32 (rw_new @ mem_new) ----------
__global__ void k_readvec(const float* __restrict__ memnew, const float* __restrict__ rwnew,
                          float* __restrict__ out) {
  const int lane = threadIdx.x;            // one wave32 per block
  const int b = blockIdx.x >> 2;
  const int wt = blockIdx.x & 3;
  const int w0 = wt * 16;
  const int row = lane & 15;
  const bool hi = lane >= 16;
  v8f acc = {};
  for (int k0 = 0; k0 < N_; k0 += 4) {
    const int ka = k0 + (hi ? 2 : 0);
    v2f a, bv;
    if (row < R_) { a.x = rwnew[(b * R_ + row) * N_ + ka]; a.y = rwnew[(b * R_ + row) * N_ + ka + 1]; }
    else          { a.x = 0.0f; a.y = 0.0f; }
    bv.x = memnew[((size_t)b * N_ + ka)     * WC_ + w0 + row];
    bv.y = memnew[((size_t)b * N_ + ka + 1) * WC_ + w0 + row];
    acc = wmma_f32_16x16x4(a, bv, acc);
  }
  if (lane < 16) {
    for (int v = 0; v < R_; ++v)
      out[OUT_RV + (size_t)(b * R_ + v) * WC_ + w0 + lane] = acc[v];
  }
}

// -------------------- launcher --------------------
extern "C" void kernel_launch(void* const* d_in, const int* in_sizes, int n_in,
                              void* d_out, int out_size, void* d_ws, size_t ws_size,
                              hipStream_t stream) {
  const float* x       = (const float*)d_in[0];
  const float* memory  = (const float*)d_in[1];
  const float* linkm   = (const float*)d_in[2];
  const float* prec    = (const float*)d_in[3];
  const float* rw_old  = (const float*)d_in[4];
  const float* wwr     = (const float*)d_in[5];
  const float* uvec    = (const float*)d_in[6];
  float* out = (float*)d_out;
  float* ws  = (float*)d_ws;

  // 1. pack weights + biases
  k_pack<<<(CTRLP_ * IN_ + 255) / 256, 256, 0, stream>>>(
      (const float*)d_in[7],  (const float*)d_in[8],  (const float*)d_in[9],  (const float*)d_in[10],
      (const float*)d_in[11], (const float*)d_in[12], (const float*)d_in[13], (const float*)d_in[14],
      (const float*)d_in[15], (const float*)d_in[16], (const float*)d_in[17], (const float*)d_in[18],
      (const float*)d_in[19], (const float*)d_in[20], (const float*)d_in[21], (const float*)d_in[22],
      (const float*)d_in[23], (const float*)d_in[24], (const float*)d_in[25], (const float*)d_in[26],
      ws);
  // 2. zero bwd accumulator
  k_zero<<<(B_ * R_ * N_ + 255) / 256, 256, 0, stream>>>(ws + WS_BWD, B_ * R_ * N_);
  // 3. controller GEMM (WMMA f32), 4. activations
  k_ctrl_gemm<<<CTRLP_ / 16, 32, 0, stream>>>(x, ws);
  k_act<<<(B_ * CTRL_ + 255) / 256, 256, 0, stream>>>(ws);
  // 5. usage (final output)
  k_usage<<<(B_ * N_ + 255) / 256, 256, 0, stream>>>(uvec, wwr, rw_old, ws, out + OUT_USAGE);
  // 6. old memory norms, 7. write content sim, 8. softmax -> wcw
  k_rownorm<<<(B_ * N_) / 8, 256, 0, stream>>>(memory, ws + WS_MNO, B_ * N_);
  k_wcw_sim<<<(B_ * N_ + 255) / 256, 256, 0, stream>>>(memory, ws);
  k_softmax<<<B_, 256, 0, stream>>>(ws + WS_WCWS, ws + WS_WCW, N_);
  // 9. allocation weighting
  k_alloc<<<B_, 256, 0, stream>>>(out + OUT_USAGE, ws);
  // 10. write weighting + precedence
  k_ww_prec<<<B_, 256, 0, stream>>>(prec, ws, out);
  // 11. memory update + new norms
  k_memnew<<<(B_ * N_ * WC_) / 256, 256, 0, stream>>>(memory, ws, out);
  k_rownorm<<<(B_ * N_) / 8, 256, 0, stream>>>(out + OUT_MEM, ws + WS_MNN, B_ * N_);
  // 12. read content sim (WMMA) + softmax -> rcw
  k_rcw_sim<<<B_ * (N_ / 16), 32, 0, stream>>>(out + OUT_MEM, ws);
  k_softmax<<<B_ * R_, 256, 0, stream>>>(ws + WS_RCWS, ws + WS_RCW, N_);
  // 13. fused link + fwd + bwd (bandwidth-bound core)
  k_link<<<B_ * 32, 256, 0, stream>>>(linkm, prec, rw_old, ws, out);
  // 14. read weighting combine
  k_rwnew<<<(B_ * R_ * N_) / 256, 256, 0, stream>>>(ws, out);
  // 15. read vectors (WMMA)
  k_readvec<<<B_ * (WC_ / 16), 32, 0, stream>>>(out + OUT_MEM, out + OUT_RW, out);
}